// MutexLoss_32401233281619
// MI455X (gfx1250) — compile-verified
//
#include <hip/hip_runtime.h>

typedef __attribute__((ext_vector_type(2))) float v2f;
typedef __attribute__((ext_vector_type(4))) float v4f;
typedef __attribute__((ext_vector_type(8))) float v8f;

constexpr int B = 32, P = 32, S = 512;
constexpr int PS  = P * S;     // 16384
constexpr int NPB = B * P;     // 1024

// workspace layout (float offsets)
constexpr int OFF_VOL   = 0;        // NPB floats
constexpr int OFF_AROWS = 1024;     // NPB * 3 * 4 floats (3 rows of (r0,r1,r2,t) per box)
constexpr int OFF_PTS   = 16384;    // B * PS * 4 floats (x,y,z,1 per generated point)

// ---------------------------------------------------------------------------
// Pass 1a: per (b,p) rotation rows for WMMA A-matrix, and vol = prod(shape)*IOU
// ---------------------------------------------------------------------------
__global__ void prep_kernel(const float* __restrict__ shape,
                            const float* __restrict__ trans,
                            const float* __restrict__ quat,
                            const float* __restrict__ iou,
                            float* __restrict__ ws) {
  int i = blockIdx.x * blockDim.x + threadIdx.x;
  if (i >= NPB) return;
  float w = quat[i*4+0], x = quat[i*4+1], y = quat[i*4+2], z = quat[i*4+3];
  float inv = rsqrtf(w*w + x*x + y*y + z*z);
  w *= inv; x *= inv; y *= inv; z *= inv;
  // R such that quat_rotate(q, v) == R v
  float R00 = 1.f-2.f*(y*y+z*z), R01 = 2.f*(x*y-w*z), R02 = 2.f*(x*z+w*y);
  float R10 = 2.f*(x*y+w*z), R11 = 1.f-2.f*(x*x+z*z), R12 = 2.f*(y*z-w*x);
  float R20 = 2.f*(x*z-w*y), R21 = 2.f*(y*z+w*x), R22 = 1.f-2.f*(x*x+y*y);
  float t0 = trans[i*3+0], t1 = trans[i*3+1], t2 = trans[i*3+2];
  float s0 = shape[i*3+0], s1 = shape[i*3+1], s2 = shape[i*3+2];
  ws[OFF_VOL + i] = s0 * s1 * s2 * iou[i];
  // row c: p_loc_c = R[0][c]*px + R[1][c]*py + R[2][c]*pz - (R[.][c]·t)
  v4f* ar = (v4f*)(ws + OFF_AROWS);
  v4f r0 = {R00, R10, R20, -(R00*t0 + R10*t1 + R20*t2)};
  v4f r1 = {R01, R11, R21, -(R01*t0 + R11*t1 + R21*t2)};
  v4f r2 = {R02, R12, R22, -(R02*t0 + R12*t1 + R22*t2)};
  ar[i*3+0] = r0; ar[i*3+1] = r1; ar[i*3+2] = r2;
}

// ---------------------------------------------------------------------------
// Pass 1b: generate pointRef[b, p*S+s] = R*( (2c-1)*shape ) + t, stored (x,y,z,1)
// ---------------------------------------------------------------------------
__global__ void points_kernel(const float* __restrict__ shape,
                              const float* __restrict__ trans,
                              const float* __restrict__ quat,
                              const float* __restrict__ coef,
                              float* __restrict__ ws) {
  int idx = blockIdx.x * blockDim.x + threadIdx.x;  // b*P*S + p*S + s
  if (idx >= B * P * S) return;
  int i = idx / S;  // b*P + p
  float qw = quat[i*4+0], qx = quat[i*4+1], qy = quat[i*4+2], qz = quat[i*4+3];
  float inv = rsqrtf(qw*qw + qx*qx + qy*qy + qz*qz);
  qw *= inv; qx *= inv; qy *= inv; qz *= inv;
  float vx = (2.f*coef[idx*3+0] - 1.f) * shape[i*3+0];
  float vy = (2.f*coef[idx*3+1] - 1.f) * shape[i*3+1];
  float vz = (2.f*coef[idx*3+2] - 1.f) * shape[i*3+2];
  float ux = qy*vz - qz*vy, uy = qz*vx - qx*vz, uz = qx*vy - qy*vx;   // qv x v
  float wx = qy*uz - qz*uy, wy = qz*ux - qx*uz, wz = qx*uy - qy*ux;   // qv x uv
  float px = vx + 2.f*(qw*ux + wx) + trans[i*3+0];
  float py = vy + 2.f*(qw*uy + wy) + trans[i*3+1];
  float pz = vz + 2.f*(qw*uz + wz) + trans[i*3+2];
  v4f pt = {px, py, pz, 1.f};
  ((v4f*)(ws + OFF_PTS))[idx] = pt;
}

// ---------------------------------------------------------------------------
// Pass 2: tsdfOut via WMMA. One wave handles (b, group of 4 boxes, 128 points).
// A (16x4): rows 0-2/3-5 = boxes pg*4+0/1, rows 8-10/11-13 = boxes pg*4+2/3.
// B (4x16): columns = 16 homogeneous points. D gives p_loc; lanes 0-15 hold
// boxes {+0,+1}, lanes 16-31 hold {+2,+3}, each with 3 coords in d[0..2]/d[3..5].
// Output stores are non-temporal: 67 MB written once, never re-read.
// ---------------------------------------------------------------------------
__global__ void __launch_bounds__(256) tsdf_kernel(const float* __restrict__ shape,
                                                   const float* __restrict__ iou,
                                                   const float* __restrict__ ws,
                                                   float* __restrict__ out) {
  const int lane = threadIdx.x & 31;
  const int wv   = (blockIdx.x * blockDim.x + threadIdx.x) >> 5;  // 0..32767
  const int jt = wv & 127;          // 128-point super-tile index
  const int pg = (wv >> 7) & 7;     // box group (4 boxes)
  const int b  = wv >> 10;          // batch
  const int ln = lane & 15;
  const int lh = lane >> 4;

  // ---- A fragment: lane ln = row M, lane-half selects K pair {0,1}/{2,3}
  v2f afrag = {0.f, 0.f};
  {
    const int m = ln;
    if ((m & 7) < 6) {
      int pl  = (m >> 3) * 2 + (m & 7) / 3;
      int c   = (m & 7) % 3;
      int box = pg * 4 + pl;
      afrag = *(const v2f*)(ws + OFF_AROWS + ((b * P + box) * 3 + c) * 4 + lh * 2);
    }
  }

  // ---- per-lane output boxes
  const int pA = pg * 4 + lh * 2;
  const int pB = pA + 1;
  const int iA = b * P + pA, iB = b * P + pB;
  const float sA0 = shape[iA*3], sA1 = shape[iA*3+1], sA2 = shape[iA*3+2], ioA = iou[iA];
  const float sB0 = shape[iB*3], sB1 = shape[iB*3+1], sB2 = shape[iB*3+2], ioB = iou[iB];

  const float* pts  = ws + OFF_PTS + (size_t)b * PS * 4;
  float* outA = out + (size_t)iA * PS;
  float* outB = out + (size_t)iB * PS;

  for (int it = 0; it < 8; ++it) {
    const int jb = jt * 128 + it * 16;
    const int j  = jb + ln;
    // B fragment: lane ln = column N (point j), lane-half selects K pair:
    // lanes 0-15 -> (x,y), lanes 16-31 -> (z,1)
    v2f bfrag = *(const v2f*)(pts + (size_t)j * 4 + lh * 2);
    v8f acc = {};
    acc = __builtin_amdgcn_wmma_f32_16x16x4_f32(
        /*neg_a=*/false, afrag, /*neg_b=*/false, bfrag,
        /*c_mod=*/(short)0, acc, /*reuse_a=*/false, /*reuse_b=*/false);

    float a0 = fmaxf(sA0 - fabsf(acc[0]), 0.f);
    float a1 = fmaxf(sA1 - fabsf(acc[1]), 0.f);
    float a2 = fmaxf(sA2 - fabsf(acc[2]), 0.f);
    float tA = (a0*a0 + a1*a1 + a2*a2) * ioA;
    float b0 = fmaxf(sB0 - fabsf(acc[3]), 0.f);
    float b1 = fmaxf(sB1 - fabsf(acc[4]), 0.f);
    float b2 = fmaxf(sB2 - fabsf(acc[5]), 0.f);
    float tB = (b0*b0 + b1*b1 + b2*b2) * ioB;

    const int owner = j >> 9;  // j / S
    __builtin_nontemporal_store((owner == pA) ? 0.f : tA, outA + j);
    __builtin_nontemporal_store((owner == pB) ? 0.f : tB, outB + j);
  }
}

// ---------------------------------------------------------------------------
// Pass 3: weight[b,p,j] = vol[b, j/S]; tsdfGT = 0. Streaming NT float4 writes.
// ---------------------------------------------------------------------------
__global__ void __launch_bounds__(256) weight_zero_kernel(const float* __restrict__ ws,
                                                          float* __restrict__ out) {
  size_t idx = (size_t)blockIdx.x * blockDim.x + threadIdx.x;
  size_t e = idx * 4;
  if (e >= (size_t)B * P * PS) return;
  int b = (int)(e / ((size_t)P * PS));
  int j = (int)(e % PS);
  float v = ws[OFF_VOL + b * P + (j >> 9)];
  v4f wv = {v, v, v, v};
  v4f zv = {0.f, 0.f, 0.f, 0.f};
  __builtin_nontemporal_store(wv, (v4f*)(out + (size_t)B * P * PS + e));
  __builtin_nontemporal_store(zv, (v4f*)(out + (size_t)2 * B * P * PS + e));
}

extern "C" void kernel_launch(void* const* d_in, const int* in_sizes, int n_in,
                              void* d_out, int out_size, void* d_ws, size_t ws_size,
                              hipStream_t stream) {
  (void)in_sizes; (void)n_in; (void)out_size; (void)ws_size;
  const float* shape = (const float*)d_in[0];
  const float* trans = (const float*)d_in[1];
  const float* quat  = (const float*)d_in[2];
  const float* iou   = (const float*)d_in[3];
  const float* coef  = (const float*)d_in[4];
  float* ws  = (float*)d_ws;
  float* out = (float*)d_out;

  prep_kernel<<<4, 256, 0, stream>>>(shape, trans, quat, iou, ws);
  points_kernel<<<(B * P * S) / 256, 256, 0, stream>>>(shape, trans, quat, coef, ws);
  tsdf_kernel<<<4096, 256, 0, stream>>>(shape, iou, ws, out);
  weight_zero_kernel<<<(B * P * PS / 4) / 256, 256, 0, stream>>>(ws, out);
}